// AudioModelX1_88424786690499
// MI455X (gfx1250) — compile-verified
//
#include <hip/hip_runtime.h>
#include <hip/hip_bf16.h>

typedef __attribute__((ext_vector_type(2))) float v2f;
typedef __attribute__((ext_vector_type(8))) float v8f;

constexpr int cB = 16, cS = 512, cD = 1024, cINNER = 2048, cNH = 4;
constexpr int cDHM = 512, cDHS = 256, cFF = 1344, cNROW = cB * cS; // 8192

__device__ __forceinline__ float siluf(float x) { return x / (1.f + expf(-x)); }
__device__ __forceinline__ float logsigf(float x) { return fminf(x, 0.f) - log1pf(expf(-fabsf(x))); }

// ---------------------------------------------------------------------------
// Tensor Data Mover (CDNA5 TDM): async 2D tile DMA global -> LDS with row pad.
// Descriptor built per ISA cdna5 ch.8 (D# group0/group1). Device-only.
// ---------------------------------------------------------------------------
#if defined(__HIP_DEVICE_COMPILE__) && defined(__gfx1250__)
#  if __has_builtin(__builtin_amdgcn_tensor_load_to_lds) && __has_builtin(__builtin_amdgcn_s_wait_tensorcnt)
#    define USE_TDM 1
#  endif
#endif
#ifndef USE_TDM
#define USE_TDM 0
#endif

#if USE_TDM
typedef unsigned int u32x4 __attribute__((ext_vector_type(4)));
typedef int i32x4 __attribute__((ext_vector_type(4)));
typedef int i32x8 __attribute__((ext_vector_type(8)));

// Load tile_h x tile_w (fp32 elements) from gptr (row stride = row_stride elems)
// into LDS at byte offset lds_off, inserting pad_amt dwords after every 32 dwords.
// pad codes: interval 4 => 32 dwords, amount 3 => 4 dwords  (LDS row stride 36).
__device__ __forceinline__ void tdm_load_2d(unsigned lds_off, const void* gptr,
                                            unsigned tile_w, unsigned tile_h,
                                            unsigned row_stride)
{
  unsigned long long ga = (unsigned long long)(size_t)gptr;
  u32x4 g0;
  g0[0] = 1u;                                                  // count=1, user desc
  g0[1] = lds_off;                                             // lds_addr (bytes)
  g0[2] = (unsigned)ga;                                        // global_addr[31:0]
  g0[3] = (unsigned)((ga >> 32) & 0x01FFFFFFu) | (2u << 30);   // addr[56:32] | type=2
  i32x8 g1;
  g1[0] = (int)((2u << 16) | (1u << 20) | (4u << 22) | (3u << 25)); // 4B, pad on, 32dw, +4dw
  g1[1] = (int)(0xFFFFu << 16);                 // tensor_dim0 = 0x7FFFFFFF (lo16)
  g1[2] = (int)(0x7FFFu | (0xFFFFu << 16));     // tensor_dim0 hi16 | tensor_dim1 lo16
  g1[3] = (int)(0x7FFFu | (tile_w << 16));      // tensor_dim1 hi16 | tile_dim0
  g1[4] = (int)(tile_h & 0xFFFFu);              // tile_dim1, tile_dim2=0
  g1[5] = (int)row_stride;                      // tensor_dim0_stride lo32
  g1[6] = 0;
  g1[7] = 0;
  i32x4 gz = {0, 0, 0, 0};
#if __clang_major__ >= 23
  i32x8 gz8 = {0, 0, 0, 0, 0, 0, 0, 0};
  __builtin_amdgcn_tensor_load_to_lds(g0, g1, gz, gz, gz8, 0);
#else
  __builtin_amdgcn_tensor_load_to_lds(g0, g1, gz, gz, 0);
#endif
}
#endif

// ---------------------------------------------------------------------------
// Generic fp32 WMMA GEMM: C[M,N] = A[M,K] @ B, tile 64x64, 8 waves per block.
// B element (k,n) = bTrans ? W[n*ldb+k] : W[k*ldb+n].  Requires M%64==N%64==0, K%32==0.
// A tile DMA'd by the TDM (wave 0, s_wait_tensorcnt); B tile stored transposed in
// LDS so all WMMA fragments are contiguous 8B-aligned pairs (single ds_load_b64).
// ---------------------------------------------------------------------------
__global__ __launch_bounds__(256) void gemm_wmma_f32(
    const float* __restrict__ A, int lda,
    const float* __restrict__ W, int ldb, int bTrans,
    float* __restrict__ C, int ldc, int M, int N, int K)
{
  __shared__ float As[64][36];   // As[m][k]
  __shared__ float Bs[64][36];   // Bs[n][k]  (transposed tile)
  const int tid  = threadIdx.x;
  const int lane = tid & 31;
  const int wave = tid >> 5;
  const int wm   = wave & 3;   // row tile 0..3
  const int wn   = wave >> 2;  // col half 0..1
  const int l16  = lane & 15;
  const int lh   = lane >> 4;
  const int bm   = blockIdx.x * 64;
  const int bn   = blockIdx.y * 64;
  (void)M;

  v8f acc0 = {0.f,0.f,0.f,0.f,0.f,0.f,0.f,0.f};
  v8f acc1 = {0.f,0.f,0.f,0.f,0.f,0.f,0.f,0.f};

  for (int k0 = 0; k0 < K; k0 += 32) {
    if (bTrans) {
      for (int e = tid; e < 64 * 32; e += 256) {
        int n = e >> 5, kk2 = e & 31;
        Bs[n][kk2] = W[(size_t)(bn + n) * ldb + (k0 + kk2)];
      }
    } else {
      for (int e = tid; e < 32 * 64; e += 256) {
        int kk2 = e >> 6, n = e & 63;
        Bs[n][kk2] = W[(size_t)(k0 + kk2) * ldb + (bn + n)];
      }
    }
#if USE_TDM
    if (tid < 32) {   // wave 0 issues the tile DMA; EXEC-independent, per-wave counter
      tdm_load_2d((unsigned)(size_t)&As[0][0], A + (size_t)bm * lda + k0,
                  32u, 64u, (unsigned)lda);
      __builtin_amdgcn_s_wait_tensorcnt(0);
    }
#else
    for (int e = tid; e < 64 * 32; e += 256) {
      int r = e >> 5, c = e & 31;
      As[r][c] = A[(size_t)(bm + r) * lda + (k0 + c)];
    }
#endif
    __syncthreads();
#pragma unroll
    for (int kk = 0; kk < 32; kk += 4) {
      v2f a, b0, b1;
      a.x  = As[wm * 16 + l16][kk + lh * 2 + 0];
      a.y  = As[wm * 16 + l16][kk + lh * 2 + 1];
      b0.x = Bs[wn * 32 + l16][kk + lh * 2 + 0];
      b0.y = Bs[wn * 32 + l16][kk + lh * 2 + 1];
      b1.x = Bs[wn * 32 + 16 + l16][kk + lh * 2 + 0];
      b1.y = Bs[wn * 32 + 16 + l16][kk + lh * 2 + 1];
      acc0 = __builtin_amdgcn_wmma_f32_16x16x4_f32(false, a, false, b0, (short)0, acc0, false, false);
      acc1 = __builtin_amdgcn_wmma_f32_16x16x4_f32(false, a, false, b1, (short)0, acc1, false, false);
    }
    __syncthreads();
  }
#pragma unroll
  for (int r = 0; r < 8; ++r) {
    int m  = bm + wm * 16 + r + lh * 8;
    int n0 = bn + wn * 32 + l16;
    C[(size_t)m * ldc + n0]      = acc0[r];
    C[(size_t)m * ldc + n0 + 16] = acc1[r];
  }
}

// ---------------------------------------------------------------------------
// LayerNorm / multi-head group norm.  gmode=1: gamma[(row%NH)*rowlen+i]; act=1: SELU.
// ---------------------------------------------------------------------------
__global__ __launch_bounds__(256) void ln_kernel(
    const float* __restrict__ in, const float* __restrict__ g,
    float* __restrict__ out, int rowlen, int gmode, int act)
{
  __shared__ float red[256];
  const int row = blockIdx.x, tid = threadIdx.x;
  const float* x = in + (size_t)row * rowlen;
  float s = 0.f;
  for (int i = tid; i < rowlen; i += 256) s += x[i];
  red[tid] = s; __syncthreads();
  for (int o = 128; o > 0; o >>= 1) { if (tid < o) red[tid] += red[tid + o]; __syncthreads(); }
  const float mu = red[0] / rowlen;
  __syncthreads();
  float v = 0.f;
  for (int i = tid; i < rowlen; i += 256) { float d = x[i] - mu; v += d * d; }
  red[tid] = v; __syncthreads();
  for (int o = 128; o > 0; o >>= 1) { if (tid < o) red[tid] += red[tid + o]; __syncthreads(); }
  const float rs = rsqrtf(red[0] / rowlen + 1e-5f);
  const float* gp = gmode ? (g + (row % cNH) * rowlen) : g;
  for (int i = tid; i < rowlen; i += 256) {
    float val = (x[i] - mu) * rs * gp[i];
    if (act == 1) {
      const float sc = 1.0507009873554805f, al = 1.6732632423543772f;
      val = (val > 0.f) ? sc * val : sc * al * (expf(val) - 1.f);
    }
    out[(size_t)row * rowlen + i] = val;
  }
}

// Depthwise causal conv (k=4) + SiLU.
__global__ __launch_bounds__(256) void conv_silu_kernel(
    const float* __restrict__ in, int ldin, const float* __restrict__ w,
    const float* __restrict__ bias, float* __restrict__ out, int C, int total)
{
  int idx = blockIdx.x * 256 + threadIdx.x;
  if (idx >= total) return;
  int c = idx % C, bs = idx / C, s = bs % cS;
  float acc = bias[c];
  const float* wc = w + c * 4;
#pragma unroll
  for (int j = 0; j < 4; ++j) {
    int sp = s - 3 + j;
    if (sp >= 0) acc += in[(size_t)(bs - s + sp) * ldin + c] * wc[j];
  }
  out[(size_t)bs * C + c] = siluf(acc);
}

// Block-diagonal 4x4 headwise linear (q/k/v); safe in-place per thread-owned block.
__global__ __launch_bounds__(256) void hw4_kernel(
    const float* __restrict__ in, int ldin, const float* __restrict__ w,
    float* __restrict__ out, int ldout)
{
  int idx = blockIdx.x * 256 + threadIdx.x;
  if (idx >= cNROW * 512) return;
  int row = idx / 512, nb = idx % 512;
  const float* xi = in + (size_t)row * ldin + nb * 4;
  float x0 = xi[0], x1 = xi[1], x2 = xi[2], x3 = xi[3];
  const float* wb = w + nb * 16;
  float* yo = out + (size_t)row * ldout + nb * 4;
#pragma unroll
  for (int o = 0; o < 4; ++o)
    yo[o] = x0 * wb[o * 4 + 0] + x1 * wb[o * 4 + 1] + x2 * wb[o * 4 + 2] + x3 * wb[o * 4 + 3];
}

// ig/fg projections: dots of length 6144 over concat(q,k,v), N=4 each.
__global__ __launch_bounds__(256) void ifgate_kernel(
    const float* __restrict__ q, const float* __restrict__ k,
    const float* __restrict__ v, int ldv,
    const float* __restrict__ igw, const float* __restrict__ igb,
    const float* __restrict__ fgw, const float* __restrict__ fgb,
    float* __restrict__ ig, float* __restrict__ fg)
{
  __shared__ float red[256];
  const int row = blockIdx.x, tid = threadIdx.x;
  const int b = row / cS, s = row % cS;
  float aI[4] = {0, 0, 0, 0}, aF[4] = {0, 0, 0, 0};
  for (int c = tid; c < 3 * cINNER; c += 256) {
    int seg = c >> 11, cc = c & 2047;
    float xv = (seg == 0) ? q[(size_t)row * cINNER + cc]
             : (seg == 1) ? k[(size_t)row * cINNER + cc]
                          : v[(size_t)row * ldv + cc];
    const float* wi = igw + c * 4;
    const float* wf = fgw + c * 4;
#pragma unroll
    for (int h = 0; h < 4; ++h) { aI[h] += xv * wi[h]; aF[h] += xv * wf[h]; }
  }
  for (int h = 0; h < 4; ++h) {
    red[tid] = aI[h]; __syncthreads();
    for (int o = 128; o > 0; o >>= 1) { if (tid < o) red[tid] += red[tid + o]; __syncthreads(); }
    if (tid == 0) ig[(size_t)(b * cNH + h) * cS + s] = red[0] + igb[h];
    __syncthreads();
    red[tid] = aF[h]; __syncthreads();
    for (int o = 128; o > 0; o >>= 1) { if (tid < o) red[tid] += red[tid + o]; __syncthreads(); }
    if (tid == 0) fg[(size_t)(b * cNH + h) * cS + s] = red[0] + fgb[h];
    __syncthreads();
  }
}

// Inclusive prefix sum of log-sigmoid(fg) per (b,head); lfc[0]=0.
__global__ __launch_bounds__(512) void lfc_kernel(const float* __restrict__ fg, float* __restrict__ lfc)
{
  __shared__ float sc[512];
  const int bn = blockIdx.x, t = threadIdx.x;
  sc[t] = logsigf(fg[(size_t)bn * cS + t]);
  __syncthreads();
  for (int off = 1; off < 512; off <<= 1) {
    float add = (t >= off) ? sc[t - off] : 0.f;
    __syncthreads();
    sc[t] += add;
    __syncthreads();
  }
  if (t == 0) lfc[(size_t)bn * (cS + 1)] = 0.f;
  lfc[(size_t)bn * (cS + 1) + t + 1] = sc[t];
}

// ---------------------------------------------------------------------------
// Fused flash-style mLSTM attention. One block per (query-tile, head, batch).
// 8 waves each own a 64-wide slice of dh=512 (q frags in regs, out accs in regs).
// k chunk stored row-major (frag pairs contiguous); v chunk stored transposed
// (vt[d][j], stride 18) so P@v B-fragments are contiguous b64 loads too.
// ---------------------------------------------------------------------------
#define KSTR 516
#define VSTR 18
#define PSTR 18
__global__ __launch_bounds__(256) void attn_kernel(
    const float* __restrict__ q, const float* __restrict__ k,
    const float* __restrict__ v, int ldv,
    const float* __restrict__ lfc, const float* __restrict__ ig,
    float* __restrict__ out)
{
  extern __shared__ float sm[];
  float* ks     = sm;                   // 16*KSTR          (k chunk, row-major)
  float* vt     = ks + 16 * KSTR;       // 512*VSTR         (v chunk, transposed)
  float* part   = vt + 512 * VSTR;      // 8*256            (per-wave qk partials)
  float* pt     = part + 2048;          // 16*PSTR          (P tile)
  float* rowm   = pt + 16 * PSTR;       // 16
  float* rowsum = rowm + 16;            // 16
  float* scal   = rowsum + 16;          // 16

  const int qt = blockIdx.x, n = blockIdx.y, b = blockIdx.z;
  const int rowbase = qt * 16;
  const int bn = b * cNH + n;
  const int tid = threadIdx.x, lane = tid & 31, wave = tid >> 5;
  const int l16 = lane & 15, lh = lane >> 4;

  v2f qa[16];
  {
    const float qs = 0.04419417382415922f; // 512^-0.5
    const float* qrow = q + (size_t)(b * cS + rowbase + l16) * cINNER + n * cDHM + wave * 64;
#pragma unroll
    for (int s4 = 0; s4 < 16; ++s4) {
      qa[s4].x = qrow[s4 * 4 + lh * 2 + 0] * qs;
      qa[s4].y = qrow[s4 * 4 + lh * 2 + 1] * qs;
    }
  }
  if (tid < 16) { rowm[tid] = -1e30f; rowsum[tid] = 0.f; }
  const v8f z8 = {0.f,0.f,0.f,0.f,0.f,0.f,0.f,0.f};
  v8f acc[4];
#pragma unroll
  for (int t = 0; t < 4; ++t) acc[t] = z8;

  const int nchunks = qt + 1;
  for (int c = 0; c < nchunks; ++c) {
    const int jb = c * 16;
    __syncthreads();
    for (int e = tid; e < 16 * cDHM; e += 256) {
      int r = e >> 9, col = e & 511;
      ks[r * KSTR + col] = k[(size_t)(b * cS + jb + r) * cINNER + n * cDHM + col];
      vt[col * VSTR + r] = v[(size_t)(b * cS + jb + r) * (size_t)ldv + n * cDHM + col];
    }
    __syncthreads();
    // q @ k^T partial over this wave's 64-wide K slice
    v8f s = z8;
#pragma unroll
    for (int s4 = 0; s4 < 16; ++s4) {
      const int kk = wave * 64 + s4 * 4;
      v2f bb;
      bb.x = ks[l16 * KSTR + kk + lh * 2 + 0];
      bb.y = ks[l16 * KSTR + kk + lh * 2 + 1];
      s = __builtin_amdgcn_wmma_f32_16x16x4_f32(false, qa[s4], false, bb, (short)0, s, false, false);
    }
#pragma unroll
    for (int r = 0; r < 8; ++r) part[wave * 256 + (r + lh * 8) * 16 + l16] = s[r];
    __syncthreads();
    // reduce partials, apply log-D gate, streaming max + signed-sum normalizer
    if (tid < 16) {
      const int i = tid;
      const float lq = lfc[(size_t)bn * (cS + 1) + rowbase + i + 1];
      const int jmax = (jb == rowbase) ? i : 15;
      float mold = rowm[i], mnew = mold;
      float qkv[16], lv[16];
      for (int j = 0; j < 16; ++j) {
        float ss = 0.f;
        for (int w2 = 0; w2 < 8; ++w2) ss += part[w2 * 256 + i * 16 + j];
        qkv[j] = ss;
        if (j <= jmax) {
          float ld_ = lq - lfc[(size_t)bn * (cS + 1) + jb + j + 1] + ig[(size_t)bn * cS + jb + j];
          lv[j] = ld_;
          mnew = fmaxf(mnew, ld_);
        }
      }
      const float scl = expf(mold - mnew);
      float psum = 0.f;
      for (int j = 0; j < 16; ++j) {
        float p = (j <= jmax) ? qkv[j] * expf(lv[j] - mnew) : 0.f;
        pt[i * PSTR + j] = p;
        psum += p;
      }
      rowsum[i] = rowsum[i] * scl + psum;
      rowm[i] = mnew;
      scal[i] = scl;
    }
    __syncthreads();
    float sc8[8];
#pragma unroll
    for (int r = 0; r < 8; ++r) sc8[r] = scal[r + lh * 8];
#pragma unroll
    for (int t = 0; t < 4; ++t)
#pragma unroll
      for (int r = 0; r < 8; ++r) acc[t][r] *= sc8[r];
    // P @ v for this wave's 4 output col tiles
#pragma unroll
    for (int kk = 0; kk < 16; kk += 4) {
      v2f pa;
      pa.x = pt[l16 * PSTR + kk + lh * 2 + 0];
      pa.y = pt[l16 * PSTR + kk + lh * 2 + 1];
#pragma unroll
      for (int t = 0; t < 4; ++t) {
        v2f bv;
        const int nb_ = wave * 64 + t * 16 + l16;
        bv.x = vt[nb_ * VSTR + kk + lh * 2 + 0];
        bv.y = vt[nb_ * VSTR + kk + lh * 2 + 1];
        acc[t] = __builtin_amdgcn_wmma_f32_16x16x4_f32(false, pa, false, bv, (short)0, acc[t], false, false);
      }
    }
  }
  __syncthreads();
  float inv8[8];
#pragma unroll
  for (int r = 0; r < 8; ++r) {
    const int i = r + lh * 8;
    float nrm = fmaxf(fabsf(rowsum[i]), expf(-rowm[i]));
    inv8[r] = 1.f / (nrm + 1e-6f);
  }
#pragma unroll
  for (int t = 0; t < 4; ++t)
#pragma unroll
    for (int r = 0; r < 8; ++r) {
      const int srow = rowbase + r + lh * 8;
      const int d = n * cDHM + wave * 64 + t * 16 + l16;
      out[(size_t)(b * cS + srow) * cINNER + d] = acc[t][r] * inv8[r];
    }
}

// h2 = silu(z) * (h2 + skip * xconv)
__global__ __launch_bounds__(256) void gate_mlstm_kernel(
    float* __restrict__ h2, const float* __restrict__ xc,
    const float* __restrict__ up, const float* __restrict__ skip, int total)
{
  int idx = blockIdx.x * 256 + threadIdx.x;
  if (idx >= total) return;
  int row = idx / cINNER, c = idx % cINNER;
  float z = up[(size_t)row * 4096 + cINNER + c];
  h2[idx] = siluf(z) * (h2[idx] + skip[c] * xc[idx]);
}

__global__ __launch_bounds__(256) void add_kernel(float* __restrict__ a, const float* __restrict__ b, int n)
{
  int idx = blockIdx.x * 256 + threadIdx.x;
  if (idx < n) a[idx] += b[idx];
}

// sLSTM sequential scan: one block per (b,head), one thread per channel.
__global__ __launch_bounds__(256) void slstm_scan_kernel(
    const float* __restrict__ wx, const float* __restrict__ Rw,
    const float* __restrict__ Rb, float* __restrict__ y)
{
  const int bn = blockIdx.x, b = bn / cNH, n = bn % cNH;
  const int o = threadIdx.x;
  __shared__ float hl[cDHS];
  float cs = 0.f, ns = 0.f, ms = 0.f;
  hl[o] = 0.f;
  __syncthreads();
  const float* rwI = Rw + ((size_t)(n * 4 + 0) * cDHS + o) * cDHS;
  const float* rwF = Rw + ((size_t)(n * 4 + 1) * cDHS + o) * cDHS;
  const float* rwZ = Rw + ((size_t)(n * 4 + 2) * cDHS + o) * cDHS;
  const float* rwO = Rw + ((size_t)(n * 4 + 3) * cDHS + o) * cDHS;
  const float rbI = Rb[(n * 4 + 0) * cDHS + o], rbF = Rb[(n * 4 + 1) * cDHS + o];
  const float rbZ = Rb[(n * 4 + 2) * cDHS + o], rbO = Rb[(n * 4 + 3) * cDHS + o];
  for (int t = 0; t < cS; ++t) {
    const float* wrow = wx + (size_t)(b * cS + t) * 4096 + n * 1024;
    float gi = wrow[0 * cDHS + o] + rbI;
    float gf = wrow[1 * cDHS + o] + rbF;
    float gz = wrow[2 * cDHS + o] + rbZ;
    float go = wrow[3 * cDHS + o] + rbO;
    for (int i = 0; i < cDHS; ++i) {
      float hp = hl[i];
      gi += hp * rwI[i]; gf += hp * rwF[i]; gz += hp * rwZ[i]; go += hp * rwO[i];
    }
    float logfpm = ms + logsigf(gf);
    float mnew = fmaxf(gi, logfpm);
    float igate = expf(gi - mnew);
    float fgate = expf(logfpm - mnew);
    cs = fgate * cs + igate * tanhf(gz);
    ns = fgate * ns + igate;
    float hnew = (1.f / (1.f + expf(-go))) * (cs / ns);
    ms = mnew;
    __syncthreads();
    hl[o] = hnew;
    y[(size_t)(b * cS + t) * cD + n * cDHS + o] = hnew;
    __syncthreads();
  }
}

// FFN: prod = gelu(gate) * up
__global__ __launch_bounds__(256) void gelu_gate_kernel(
    const float* __restrict__ up2, float* __restrict__ prod, int total)
{
  int idx = blockIdx.x * 256 + threadIdx.x;
  if (idx >= total) return;
  int row = idx / cFF, c = idx % cFF;
  float g = up2[(size_t)row * (2 * cFF) + c];
  float u = up2[(size_t)row * (2 * cFF) + cFF + c];
  prod[idx] = 0.5f * g * (1.f + erff(g * 0.7071067811865476f)) * u;
}

__global__ __launch_bounds__(256) void mean_pool_kernel(const float* __restrict__ t0, float* __restrict__ pool)
{
  int idx = blockIdx.x * 256 + threadIdx.x;
  if (idx >= cB * cD) return;
  int b = idx / cD, d = idx % cD;
  float s = 0.f;
  for (int t = 0; t < cS; ++t) s += t0[(size_t)(b * cS + t) * cD + d];
  pool[idx] = s / (float)cS;
}

__global__ __launch_bounds__(256) void heads_kernel(
    const float* __restrict__ pool, const float* __restrict__ ew, const float* __restrict__ eb,
    const float* __restrict__ sw, const float* __restrict__ sb, float* __restrict__ outp)
{
  __shared__ float red[256];
  const int bid = blockIdx.x, b = bid / 10, j = bid % 10, tid = threadIdx.x;
  float s = 0.f;
  for (int kx = tid; kx < cD; kx += 256) {
    float x = pool[b * cD + kx];
    s += x * ((j < 7) ? ew[kx * 7 + j] : sw[kx * 3 + (j - 7)]);
  }
  red[tid] = s; __syncthreads();
  for (int o = 128; o > 0; o >>= 1) { if (tid < o) red[tid] += red[tid + o]; __syncthreads(); }
  if (tid == 0) {
    if (j < 7) outp[b * 7 + j] = red[0] + eb[j];
    else       outp[112 + b * 3 + (j - 7)] = red[0] + sb[j - 7];
  }
}

// ---------------------------------------------------------------------------
// Host orchestration
// ---------------------------------------------------------------------------
static void run_gemm(hipStream_t st, const float* A, int lda, const float* W, int ldb,
                     int bT, float* C, int ldc, int M, int N, int K)
{
  dim3 g(M / 64, N / 64);
  gemm_wmma_f32<<<g, 256, 0, st>>>(A, lda, W, ldb, bT, C, ldc, M, N, K);
}

static void run_mlstm(hipStream_t st, const float* const* P,
                      float* h, float* t0, float* up, float* xc, float* qb, float* kb,
                      float* h2, float* yb, float* igb, float* fgb, float* lfcb)
{
  const int ATTN_SMEM = (16 * KSTR + 512 * VSTR + 2048 + 16 * PSTR + 48) * 4;
  ln_kernel<<<cNROW, 256, 0, st>>>(h, P[0], t0, cD, 0, 0);
  run_gemm(st, t0, cD, P[1], 4096, 0, up, 4096, cNROW, 4096, cD);
  conv_silu_kernel<<<(cNROW * cINNER) / 256, 256, 0, st>>>(up, 4096, P[2], P[3], xc, cINNER, cNROW * cINNER);
  hw4_kernel<<<(cNROW * 512) / 256, 256, 0, st>>>(xc, cINNER, P[4], qb, cINNER);
  hw4_kernel<<<(cNROW * 512) / 256, 256, 0, st>>>(xc, cINNER, P[5], kb, cINNER);
  hw4_kernel<<<(cNROW * 512) / 256, 256, 0, st>>>(up, 4096, P[6], up, 4096);   // v in place over x_m
  ifgate_kernel<<<cNROW, 256, 0, st>>>(qb, kb, up, 4096, P[7], P[8], P[9], P[10], igb, fgb);
  lfc_kernel<<<cB * cNH, 512, 0, st>>>(fgb, lfcb);
  attn_kernel<<<dim3(cS / 16, cNH, cB), 256, ATTN_SMEM, st>>>(qb, kb, up, 4096, lfcb, igb, h2);
  ln_kernel<<<cNROW * cNH, 256, 0, st>>>(h2, P[12], h2, cDHM, 1, 0);
  gate_mlstm_kernel<<<(cNROW * cINNER) / 256, 256, 0, st>>>(h2, xc, up, P[11], cNROW * cINNER);
  run_gemm(st, h2, cINNER, P[13], cD, 0, yb, cD, cNROW, cD, cINNER);
  add_kernel<<<(cNROW * cD) / 256, 256, 0, st>>>(h, yb, cNROW * cD);
}

static void run_slstm_ffn(hipStream_t st, const float* const* P,
                          float* h, float* t0, float* up, float* xc, float* h2, float* yb)
{
  ln_kernel<<<cNROW, 256, 0, st>>>(h, P[0], t0, cD, 0, 0);
  conv_silu_kernel<<<(cNROW * cD) / 256, 256, 0, st>>>(t0, cD, P[1], P[2], xc, cD, cNROW * cD);
  const float* gw[4]  = {P[3], P[4], P[5], P[6]};
  const float* gin[4] = {xc, xc, t0, t0};
  for (int n = 0; n < 4; ++n)
    for (int g = 0; g < 4; ++g)
      run_gemm(st, gin[g] + n * cDHS, cD, gw[g] + n * cDHS * cDHS, cDHS, 1,
               up + n * 1024 + g * cDHS, 4096, cNROW, cDHS, cDHS);
  slstm_scan_kernel<<<cB * cNH, cDHS, 0, st>>>(up, P[7], P[8], h2);
  ln_kernel<<<cNROW * cNH, 256, 0, st>>>(h2, P[9], h2, cDHS, 1, 0);
  add_kernel<<<(cNROW * cD) / 256, 256, 0, st>>>(h, h2, cNROW * cD);
  // gated FFN
  ln_kernel<<<cNROW, 256, 0, st>>>(h, P[10], t0, cD, 0, 0);
  run_gemm(st, t0, cD, P[11], 2 * cFF, 0, up, 2 * cFF, cNROW, 2 * cFF, cD);
  gelu_gate_kernel<<<(cNROW * cFF) / 256, 256, 0, st>>>(up, h2, cNROW * cFF);
  run_gemm(st, h2, cFF, P[12], cD, 0, yb, cD, cNROW, cD, cFF);
  add_kernel<<<(cNROW * cD) / 256, 256, 0, st>>>(h, yb, cNROW * cD);
}

extern "C" void kernel_launch(void* const* d_in, const int* in_sizes, int n_in,
                              void* d_out, int out_size, void* d_ws, size_t ws_size,
                              hipStream_t stream)
{
  (void)in_sizes; (void)n_in; (void)out_size; (void)ws_size;
  const float* x = (const float*)d_in[0];
  const float* p0[14]; for (int i = 0; i < 14; ++i) p0[i] = (const float*)d_in[1 + i];
  const float* p1[13]; for (int i = 0; i < 13; ++i) p1[i] = (const float*)d_in[15 + i];
  const float* p2[14]; for (int i = 0; i < 14; ++i) p2[i] = (const float*)d_in[28 + i];
  const float* post_w = (const float*)d_in[42];
  const float* emo_w = (const float*)d_in[43];
  const float* emo_b = (const float*)d_in[44];
  const float* sen_w = (const float*)d_in[45];
  const float* sen_b = (const float*)d_in[46];

  float* ws = (float*)d_ws;   // needs ~504 MB
  float* h    = ws;                       // 8192*1024
  float* t0   = ws + 8388608;             // 8192*1024
  float* up   = ws + 16777216;            // 8192*4096
  float* xc   = ws + 50331648;            // 8192*2048
  float* qb   = ws + 67108864;            // 8192*2048
  float* kb   = ws + 83886080;            // 8192*2048
  float* h2   = ws + 100663296;           // 8192*2048
  float* yb   = ws + 117440512;           // 8192*1024
  float* igb  = ws + 125829120;           // 32768
  float* fgb  = igb + 32768;              // 32768
  float* lfcb = fgb + 32768;              // 32832
  float* pool = lfcb + 32832;             // 16384

  hipMemcpyAsync(h, x, (size_t)cNROW * cD * sizeof(float), hipMemcpyDeviceToDevice, stream);

  run_mlstm(stream, p0, h, t0, up, xc, qb, kb, h2, yb, igb, fgb, lfcb);
  run_slstm_ffn(stream, p1, h, t0, up, xc, h2, yb);
  run_mlstm(stream, p2, h, t0, up, xc, qb, kb, h2, yb, igb, fgb, lfcb);

  ln_kernel<<<cNROW, 256, 0, stream>>>(h, post_w, t0, cD, 0, 1);   // LN + SELU
  mean_pool_kernel<<<(cB * cD) / 256, 256, 0, stream>>>(t0, pool);
  heads_kernel<<<cB * 10, 256, 0, stream>>>(pool, emo_w, emo_b, sen_w, sen_b, (float*)d_out);
}